// DiVeQSchemaStore_19774029430968
// MI455X (gfx1250) — compile-verified
//
#include <hip/hip_runtime.h>

typedef __attribute__((ext_vector_type(2))) float v2f;
typedef __attribute__((ext_vector_type(4))) float v4f;
typedef __attribute__((ext_vector_type(8))) float v8f;
typedef __attribute__((ext_vector_type(4))) int   v4i;

#define D 128
#define NSEG 255
#define WAVES 4
#define ROWS_PER_WAVE 16
#define ROWS_PER_BLOCK (WAVES * ROWS_PER_WAVE)
#define TILES 2          // row-tiles per wave; next tile prefetched async into LDS
#define LDS_STRIDE 132   // pad 128 -> 132 floats: conflict-free column reads
#define KSTEPS (D / 4)   // 32 wmma k-steps of depth 4
#define EPS_F 1e-8f

#if __has_builtin(__builtin_amdgcn_global_load_async_to_lds_b128)
#define HAVE_ASYNC_LDS 1
#else
#define HAVE_ASYNC_LDS 0
#endif

// Builtin signature (from hipcc diagnostic): first param is v4i* in
// address_space(1) (printed as "__device__"), second is the LDS side.
typedef __attribute__((address_space(1))) v4i* g4p_t;
typedef __attribute__((address_space(3))) v4i* l4p_t;

__device__ __forceinline__ void wait_asynccnt0() {
#if __has_builtin(__builtin_amdgcn_s_wait_asynccnt)
  __builtin_amdgcn_s_wait_asynccnt(0);
#else
  asm volatile("s_wait_asynccnt 0x0" ::: "memory");
#endif
}

__device__ __forceinline__ void wait_dscnt0() {
#if __has_builtin(__builtin_amdgcn_s_wait_dscnt)
  __builtin_amdgcn_s_wait_dscnt(0);
#else
  asm volatile("s_wait_dscnt 0x0" ::: "memory");
#endif
}

// Stage a 16x128 f32 tile: HBM -> LDS. Async (no VGPR round-trip) when available.
__device__ __forceinline__ void stage_tile(const float* __restrict__ src,
                                           float* dstLds, int lane) {
#if HAVE_ASYNC_LDS
  #pragma unroll
  for (int r = 0; r < ROWS_PER_WAVE; ++r) {
    __builtin_amdgcn_global_load_async_to_lds_b128(
        (g4p_t)(src + (size_t)r * D + lane * 4),
        (l4p_t)(dstLds + r * LDS_STRIDE + lane * 4),
        /*offset=*/0, /*cpol=*/0);
  }
#else
  #pragma unroll
  for (int r = 0; r < ROWS_PER_WAVE; ++r) {
    v4f v = *(const v4f*)(src + (size_t)r * D + lane * 4);
    *(v4f*)(dstLds + r * LDS_STRIDE + lane * 4) = v;
  }
#endif
}

// ---------------------------------------------------------------------------
// Pass 1: per-segment stats  ss = ||seg||^2 + eps, cds = ci.seg, cc = ||ci||^2
// Also zeroes usage histogram and loss (harness does not re-poison d_out).
// ---------------------------------------------------------------------------
__global__ __launch_bounds__(256)
void seg_stats_kernel(const float* __restrict__ cb,
                      float* __restrict__ ss, float* __restrict__ cds,
                      float* __restrict__ cc,
                      int* __restrict__ usage, float* __restrict__ loss) {
  int n = threadIdx.x;
  usage[n] = 0;
  if (n == 0) *loss = 0.0f;
  if (n < NSEG) {
    float a_ss = 0.f, a_cds = 0.f, a_cc = 0.f;
    for (int d0 = 0; d0 < D; d0 += 4) {
      v4f c0 = *(const v4f*)(cb + (size_t)n * D + d0);
      v4f c1 = *(const v4f*)(cb + (size_t)(n + 1) * D + d0);
      v4f s = c1 - c0;
      a_ss  += s.x * s.x + s.y * s.y + s.z * s.z + s.w * s.w;
      a_cds += c0.x * s.x + c0.y * s.y + c0.z * s.z + c0.w * s.w;
      a_cc  += c0.x * c0.x + c0.y * c0.y + c0.z * c0.z + c0.w * c0.w;
    }
    ss[n] = a_ss + EPS_F;
    cds[n] = a_cds;
    cc[n] = a_cc;
  } else {           // pad entry 255 (masked later) with benign values
    ss[n] = 1.0f; cds[n] = 0.0f; cc[n] = 0.0f;
  }
}

// ---------------------------------------------------------------------------
// Pass 2: fused  G = z @ C^T  (WMMA f32 16x16x4)  ->  segment distances ->
// argmin -> z_q gather + outputs.  Each wave processes TILES row-tiles with
// async HBM->LDS prefetch of the next tile overlapped with WMMA compute.
// ---------------------------------------------------------------------------
__global__ __launch_bounds__(WAVES * 32)
void diveq_kernel(const float* __restrict__ bt, const float* __restrict__ cb,
                  const float* __restrict__ ss_g, const float* __restrict__ cds_g,
                  const float* __restrict__ cc_g,
                  float* __restrict__ zq, int* __restrict__ idx_out,
                  float* __restrict__ lam_out, float* __restrict__ dist_out,
                  int* __restrict__ usage) {
  __shared__ float sZ[WAVES][ROWS_PER_WAVE * LDS_STRIDE];
  __shared__ float sSS[256], sCDS[256], sCC[256];

  const int tid  = threadIdx.x;
  const int lane = tid & 31;
  const int wave = tid >> 5;
  const int half = lane >> 4;          // 0: lanes 0-15, 1: lanes 16-31
  const int ncol = lane & 15;
  const int m    = lane & 15;
  const int koff = half * 2;

  // Stage per-segment stats into LDS (block-cooperative).
  for (int i = tid; i < 256; i += blockDim.x) {
    sSS[i] = ss_g[i]; sCDS[i] = cds_g[i]; sCC[i] = cc_g[i];
  }

  const int blockBase = blockIdx.x * ROWS_PER_BLOCK * TILES;
  float* myBuf = &sZ[wave][0];

  // Kick off tile 0 staging before the stats barrier.
  stage_tile(bt + (size_t)(blockBase + wave * ROWS_PER_WAVE) * D, myBuf, lane);
  wait_asynccnt0();
  __syncthreads();

  for (int t = 0; t < TILES; ++t) {
    const int rowbase = blockBase + t * ROWS_PER_BLOCK + wave * ROWS_PER_WAVE;

    // Build 32 A-fragments (16x4 f32 layout: lane m=l&15, VGPR pair holds
    // K = 4k + 2*(l>>4) .. +1).  Also accumulate ||z||^2 per row.
    v2f a[KSTEPS];
    float zzpart = 0.f;
    {
      const float* zrow = myBuf + m * LDS_STRIDE;
      #pragma unroll
      for (int k = 0; k < KSTEPS; ++k) {
        v2f v = *(const v2f*)(zrow + k * 4 + koff);
        a[k] = v;
        zzpart += v.x * v.x + v.y * v.y;
      }
    }

    // Buffer consumed -> prefetch next tile into the same buffer; the async
    // copy overlaps with the whole WMMA chunk loop below.  (ASYNC ops are
    // unordered vs DS ops, so drain DS reads first.)
    if (t + 1 < TILES) {
      wait_dscnt0();
      stage_tile(bt + (size_t)(blockBase + (t + 1) * ROWS_PER_BLOCK +
                               wave * ROWS_PER_WAVE) * D, myBuf, lane);
    }

    // lane l and l+16 each hold half the columns of row m: combine.
    float zzfull = zzpart + __shfl_xor(zzpart, 16, 32);
    // C-matrix layout: VGPR j <-> row (j + 8*half). Pre-broadcast zz per j.
    float zzrow[8];
    #pragma unroll
    for (int j = 0; j < 8; ++j)
      zzrow[j] = __shfl(zzfull, j + (half << 3), 32);

    float bestd2[8], bestt[8];
    int bestn[8];
    #pragma unroll
    for (int j = 0; j < 8; ++j) { bestd2[j] = 3.4e38f; bestt[j] = 0.f; bestn[j] = 0; }

    // Epilogue for a finished chunk: G8 holds G[:, nb..nb+15]; GN8 supplies
    // the boundary column G[:, nb+16] (its ncol==0 lanes).
    auto window = [&](int nb, const v8f& G8, const v8f& GN8, bool last) {
      const int n = nb + ncol;
      const float ssv = sSS[n], cdsv = sCDS[n], ccv = sCC[n];
      const bool valid = (!last) || (ncol < 15);   // segment 255 does not exist
      #pragma unroll
      for (int j = 0; j < 8; ++j) {
        float G  = G8[j];
        float Gr = __shfl(G, lane + 1, 32);         // G[:, n+1] within chunk
        float Gb = __shfl(GN8[j], lane & 16, 32);   // boundary from next chunk
        float G1 = (ncol == 15) ? Gb : Gr;
        float u  = G1 - G - cdsv;
        float tt = fminf(fmaxf(u / ssv, 0.f), 1.f);
        float zc2 = zzrow[j] - 2.f * G + ccv;
        float d2  = zc2 - 2.f * tt * u + tt * tt * ssv;
        if (valid && (d2 < bestd2[j])) { bestd2[j] = d2; bestn[j] = n; bestt[j] = tt; }
      }
    };

    // Main loop: 16 N-chunks of the single GEMM G = z @ C^T.
    v8f accPrev;
    for (int c = 0; c < 16; ++c) {
      v8f acc = {};
      const int n = c * 16 + ncol;                 // B-frag: lane holds col n
      const float* brow = cb + (size_t)n * D + koff;
      #pragma unroll
      for (int k = 0; k < KSTEPS; ++k) {
        v2f b = *(const v2f*)(brow + k * 4);
        acc = __builtin_amdgcn_wmma_f32_16x16x4_f32(
            /*neg_a=*/false, a[k], /*neg_b=*/false, b,
            /*c_mod=*/(short)0, acc, /*reuse_a=*/false, /*reuse_b=*/false);
      }
      if (c > 0) window((c - 1) * 16, accPrev, acc, false);
      accPrev = acc;
    }
    window(15 * 16, accPrev, accPrev, true);       // tail: ncol==15 masked

    // Reduce (d2, n, t) across the 16 lanes sharing each row; tie-break
    // lower n to match argmin semantics.
    #pragma unroll
    for (int mask = 1; mask < 16; mask <<= 1) {
      #pragma unroll
      for (int j = 0; j < 8; ++j) {
        float od = __shfl_xor(bestd2[j], mask, 32);
        int   on = __shfl_xor(bestn[j],  mask, 32);
        float ot = __shfl_xor(bestt[j],  mask, 32);
        bool take = (od < bestd2[j]) || (od == bestd2[j] && on < bestn[j]);
        if (take) { bestd2[j] = od; bestn[j] = on; bestt[j] = ot; }
      }
    }

    // Scalar outputs: lane 0 writes rows 0..7, lane 16 writes rows 8..15.
    if ((lane & 15) == 0) {
      #pragma unroll
      for (int j = 0; j < 8; ++j) {
        int r = rowbase + j + (half << 3);
        idx_out[r]  = bestn[j];
        lam_out[r]  = bestt[j];
        dist_out[r] = sqrtf(fmaxf(bestd2[j], 0.f));
        atomicAdd(&usage[bestn[j]], 1);
      }
    }

    // z_q = (1-lam)*C[idx] + lam*C[idx+1], coalesced: 32 lanes x float4/row.
    #pragma unroll
    for (int r = 0; r < ROWS_PER_WAVE; ++r) {
      const int src = (r < 8) ? 0 : 16;
      const int j = r & 7;
      int   idxr = __shfl(bestn[j], src, 32);
      float lamr = __shfl(bestt[j], src, 32);
      const float* c0 = cb + (size_t)idxr * D + lane * 4;
      v4f v0 = *(const v4f*)c0;
      v4f v1 = *(const v4f*)(c0 + D);
      v4f o;
      o.x = v0.x + lamr * (v1.x - v0.x);
      o.y = v0.y + lamr * (v1.y - v0.y);
      o.z = v0.z + lamr * (v1.z - v0.z);
      o.w = v0.w + lamr * (v1.w - v0.w);
      *(v4f*)(zq + (size_t)(rowbase + r) * D + lane * 4) = o;
    }

    // Next tile's async staging must have landed before A-frag build reads it.
    if (t + 1 < TILES) wait_asynccnt0();
  }
}

// ---------------------------------------------------------------------------
// Host launch. Output tuple layout (flat, in return order):
//   z_q[B*128] | loss[1] | indices[B] | lambdas[B] | dists[B] | usage[256]
// ---------------------------------------------------------------------------
extern "C" void kernel_launch(void* const* d_in, const int* in_sizes, int n_in,
                              void* d_out, int out_size, void* d_ws, size_t ws_size,
                              hipStream_t stream) {
  const float* bt = (const float*)d_in[0];
  const float* cb = (const float*)d_in[1];
  const int B = in_sizes[0] / D;   // 131072

  float* out = (float*)d_out;
  const size_t offLoss = (size_t)B * D;
  float* zq       = out;
  float* loss     = out + offLoss;
  int*   idx_out  = (int*)(out + offLoss + 1);
  float* lam_out  = out + offLoss + 1 + (size_t)B;
  float* dist_out = out + offLoss + 1 + 2 * (size_t)B;
  int*   usage    = (int*)(out + offLoss + 1 + 3 * (size_t)B);

  float* ws  = (float*)d_ws;
  float* ss  = ws;
  float* cds = ws + 256;
  float* cc  = ws + 512;

  seg_stats_kernel<<<1, 256, 0, stream>>>(cb, ss, cds, cc, usage, loss);

  const int blocks = B / (ROWS_PER_BLOCK * TILES);  // 131072 / 128 = 1024
  diveq_kernel<<<blocks, WAVES * 32, 0, stream>>>(
      bt, cb, ss, cds, cc, zq, idx_out, lam_out, dist_out, usage);
}